// GCNConv_42202348651103
// MI455X (gfx1250) — compile-verified
//
#include <hip/hip_runtime.h>
#include <hip/hip_bf16.h>

typedef __attribute__((ext_vector_type(2))) float v2f;
typedef __attribute__((ext_vector_type(8))) float v8f;
typedef __attribute__((ext_vector_type(4))) unsigned int u32x4;
typedef __attribute__((ext_vector_type(4))) int i32x4;
typedef __attribute__((ext_vector_type(8))) int i32x8;

#define D_FEAT 256
#define UNITS  256
#define KP     16          // K-panel staged in LDS per GEMM iteration
#define SB_STRIDE 260      // 256 + 4 pad (TDM: pad_interval=256DW, pad_amount=4DW)
#define SA_STRIDE 20       // 16 + 4 pad  (TDM: pad_interval=16DW,  pad_amount=4DW)

#if __has_builtin(__builtin_amdgcn_tensor_load_to_lds) && \
    __has_builtin(__builtin_amdgcn_s_wait_tensorcnt)
#define GCN_HAVE_TDM 1
#else
#define GCN_HAVE_TDM 0
#endif

#if GCN_HAVE_TDM
// Issue one 2D TDM tile load (global -> LDS) per CDNA5 ISA ch.8 D# layout.
// tensor dims == tile dims (tiles are always fully in-bounds by construction),
// data_size = 4 bytes, LDS padding enabled to produce bank-staggered strides.
__device__ __forceinline__ void tdm_load_2d(unsigned ldsAddr, const void* gptr,
                                            unsigned tileD0, unsigned tileD1,
                                            unsigned long long strideD0,
                                            unsigned padIntervalCode,
                                            unsigned padAmountCode) {
    const unsigned long long ga = (unsigned long long)(uintptr_t)gptr;
    u32x4 g0;
    g0[0] = 1u;                                            // count=1, user mode
    g0[1] = ldsAddr;                                       // lds_addr
    g0[2] = (unsigned)(ga & 0xFFFFFFFFull);                // global_addr[31:0]
    g0[3] = (unsigned)((ga >> 32) & 0x1FFFFFFull)          // global_addr[56:32]
          | (2u << 30);                                    // type = 2 ("image")
    i32x8 g1;
    g1[0] = (int)((2u << 16)                               // data_size = 4B
                | (1u << 20)                               // pad_enable
                | (padIntervalCode << 22)
                | (padAmountCode << 25));
    g1[1] = (int)((tileD0 & 0xFFFFu) << 16);               // tensor_dim0 lo16
    g1[2] = (int)(((tileD0 >> 16) & 0xFFFFu)               // tensor_dim0 hi16
                | ((tileD1 & 0xFFFFu) << 16));             // tensor_dim1 lo16
    g1[3] = (int)(((tileD1 >> 16) & 0xFFFFu)               // tensor_dim1 hi16
                | ((tileD0 & 0xFFFFu) << 16));             // tile_dim0
    g1[4] = (int)(tileD1 & 0xFFFFu);                       // tile_dim1 (dim2=0)
    g1[5] = (int)(strideD0 & 0xFFFFFFFFull);               // dim0_stride lo32
    g1[6] = (int)((strideD0 >> 32) & 0xFFFFull);           // dim0_stride hi16
    g1[7] = 0;
    const i32x4 z4 = {0, 0, 0, 0};
#if defined(__clang_major__) && __clang_major__ >= 23
    const i32x8 z8 = {0, 0, 0, 0, 0, 0, 0, 0};
    __builtin_amdgcn_tensor_load_to_lds(g0, g1, z4, z4, z8, 0);
#else
    __builtin_amdgcn_tensor_load_to_lds(g0, g1, z4, z4, 0);
#endif
}
#endif  // GCN_HAVE_TDM

// ===========================================================================
// GEMM: H = X @ W via V_WMMA_F32_16X16X4_F32.
// Double-buffered LDS panels; panel k+1 is DMA'd by the Tensor Data Mover
// (TENSORcnt) while the 8 waves run WMMAs on panel k. Sync fill fallback for
// toolchains without the TDM builtin and for partial row tiles.
// Block: 256 threads = 8 waves, covers 16 rows x 256 cols of H.
// ===========================================================================
__global__ void gcn_gemm_xw(const float* __restrict__ X,
                            const float* __restrict__ W,
                            float* __restrict__ H,
                            int nRows) {
    __shared__ float sA[2][16][SA_STRIDE];
    __shared__ float sB[2][KP][SB_STRIDE];

    const int tid  = threadIdx.x;
    const int wave = tid >> 5;
    const int lane = tid & 31;
    const int half = lane >> 4;
    const int l16  = lane & 15;
    const int row0 = blockIdx.x * 16;
    const int col0 = wave * 32;
    const bool full = (row0 + 16 <= nRows);    // uniform across block

    v8f c0 = {};
    v8f c1 = {};

    // ---- panel fill: TDM fast path or synchronous cooperative fallback ----
    auto fill = [&](int kp, int b) {
#if GCN_HAVE_TDM
        if (full) {
            if (wave == 0) {      // TDM ignores EXEC; one issuing wave is enough
                tdm_load_2d((unsigned)(uintptr_t)&sB[b][0][0],
                            W + (size_t)kp * UNITS,
                            /*tileD0=*/UNITS, /*tileD1=*/KP,
                            /*stride=*/UNITS, /*padI=*/7, /*padA=*/3);
                tdm_load_2d((unsigned)(uintptr_t)&sA[b][0][0],
                            X + (size_t)row0 * D_FEAT + kp,
                            /*tileD0=*/KP, /*tileD1=*/16,
                            /*stride=*/D_FEAT, /*padI=*/3, /*padA=*/3);
            }
            return;
        }
#endif
        // W panel: KP x 256 = 1024 float4, 4 per thread
        #pragma unroll
        for (int f = tid; f < KP * 64; f += 256) {
            const int r  = f >> 6;
            const int c4 = (f & 63) << 2;
            float4 wv = *(const float4*)(W + (size_t)(kp + r) * UNITS + c4);
            *(float4*)(&sB[b][r][c4]) = wv;
        }
        // X panel: 16 x KP = 64 float4
        if (tid < 64) {
            const int r  = tid >> 2;
            const int c4 = (tid & 3) << 2;
            float4 xv = {0.f, 0.f, 0.f, 0.f};
            if (full || (row0 + r) < nRows)
                xv = *(const float4*)(X + (size_t)(row0 + r) * D_FEAT + kp + c4);
            *(float4*)(&sA[b][r][c4]) = xv;
        }
    };
    auto fillWait = [&]() {
#if GCN_HAVE_TDM
        if (full && wave == 0) __builtin_amdgcn_s_wait_tensorcnt(0);
#endif
        __syncthreads();
    };

    int buf = 0;
    fill(0, 0);
    fillWait();

    for (int kp = 0; kp < D_FEAT; kp += KP) {
        const int nxt = kp + KP;
        if (nxt < D_FEAT) fill(nxt, buf ^ 1);   // async DMA of next panel

        #pragma unroll
        for (int kk = 0; kk < KP; kk += 4) {
            const int kh = kk + half * 2;
            v2f a;
            a[0] = sA[buf][l16][kh + 0];
            a[1] = sA[buf][l16][kh + 1];
            v2f b0, b1;
            b0[0] = sB[buf][kh + 0][col0 + l16];
            b0[1] = sB[buf][kh + 1][col0 + l16];
            b1[0] = sB[buf][kh + 0][col0 + 16 + l16];
            b1[1] = sB[buf][kh + 1][col0 + 16 + l16];
            c0 = __builtin_amdgcn_wmma_f32_16x16x4_f32(
                    false, a, false, b0, (short)0, c0, false, false);
            c1 = __builtin_amdgcn_wmma_f32_16x16x4_f32(
                    false, a, false, b1, (short)0, c1, false, false);
        }

        if (nxt < D_FEAT) fillWait();           // next panel landed; buf free
        buf ^= 1;
    }

    // ---- store D (C layout: VGPR j -> M=j (lanes 0-15) / M=j+8 (16-31)) ----
    if (full) {
        float* hp = H + (size_t)(row0 + half * 8) * UNITS;
        #pragma unroll
        for (int j = 0; j < 8; ++j) {
            hp[(size_t)j * UNITS + col0 + l16]      = c0[j];
            hp[(size_t)j * UNITS + col0 + 16 + l16] = c1[j];
        }
    } else {
        #pragma unroll
        for (int j = 0; j < 8; ++j) {
            const int r = row0 + j + half * 8;
            if (r < nRows) {
                H[(size_t)r * UNITS + col0 + l16]      = c0[j];
                H[(size_t)r * UNITS + col0 + 16 + l16] = c1[j];
            }
        }
    }
}

// ===========================================================================
// CSR build path (avoids the 819M-f32-atomic scatter storm)
// ===========================================================================
__global__ void gcn_zero_ints(int* __restrict__ p, int n) {
    int i = blockIdx.x * blockDim.x + threadIdx.x;
    if (i < n) p[i] = 0;
}

__global__ void gcn_degree(const int* __restrict__ eRow,
                           int* __restrict__ deg, int nEdges) {
    int i = blockIdx.x * blockDim.x + threadIdx.x;
    if (i < nEdges) atomicAdd(&deg[eRow[i]], 1);
}

// Single-workgroup exclusive scan: off[0..n] from deg[0..n-1].
__global__ void gcn_scan(const int* __restrict__ deg,
                         int* __restrict__ off, int n) {
    __shared__ int s[1024];
    __shared__ int sCarry;
    const int tid = threadIdx.x;
    if (tid == 0) sCarry = 0;
    __syncthreads();
    for (int base = 0; base < n; base += 1024) {
        const int idx = base + tid;
        const int v = (idx < n) ? deg[idx] : 0;
        s[tid] = v;
        __syncthreads();
        #pragma unroll
        for (int d = 1; d < 1024; d <<= 1) {
            int t = (tid >= d) ? s[tid - d] : 0;
            __syncthreads();
            s[tid] += t;
            __syncthreads();
        }
        if (idx < n) off[idx] = sCarry + s[tid] - v;   // exclusive
        __syncthreads();
        if (tid == 1023) sCarry += s[1023];
        __syncthreads();
    }
    if (tid == 0) off[n] = sCarry;
}

__global__ void gcn_bin(const int* __restrict__ eRow,
                        const int* __restrict__ eCol,
                        const float* __restrict__ eVal,
                        const int* __restrict__ off,
                        int* __restrict__ cursor,
                        int* __restrict__ binCol,
                        float* __restrict__ binVal, int nEdges) {
    int i = blockIdx.x * blockDim.x + threadIdx.x;
    if (i >= nEdges) return;
    const int r = eRow[i];
    const int pos = off[r] + atomicAdd(&cursor[r], 1);
    binCol[pos] = eCol[i];
    binVal[pos] = eVal[i];
}

// Row-parallel SpMM: one wave32 per node, register accumulation, no f32
// atomics, fused bias+ReLU, out written exactly once. H rows stream from L2.
__global__ void gcn_spmm_rows(const int* __restrict__ off,
                              const int* __restrict__ binCol,
                              const float* __restrict__ binVal,
                              const float* __restrict__ H,
                              const float* __restrict__ bias,
                              float* __restrict__ out, int nNodes) {
    const int node = (blockIdx.x * blockDim.x + threadIdx.x) >> 5;
    const int lane = threadIdx.x & 31;
    if (node >= nNodes) return;

    const int e0 = off[node];
    const int e1 = off[node + 1];

    float4 acc0 = {0.f, 0.f, 0.f, 0.f};
    float4 acc1 = {0.f, 0.f, 0.f, 0.f};

    for (int e = e0; e < e1; ++e) {
        const int   c = binCol[e];
        const float v = binVal[e];
        if (e + 1 < e1) {   // prefetch next gathered row (global_prefetch_b8)
            const int cn = binCol[e + 1];
            __builtin_prefetch(H + (size_t)cn * UNITS + lane * 8, 0, 0);
        }
        const float4* hp = (const float4*)(H + (size_t)c * UNITS);
        const float4 h0 = hp[lane];
        const float4 h1 = hp[lane + 32];
        acc0.x += v * h0.x; acc0.y += v * h0.y;
        acc0.z += v * h0.z; acc0.w += v * h0.w;
        acc1.x += v * h1.x; acc1.y += v * h1.y;
        acc1.z += v * h1.z; acc1.w += v * h1.w;
    }

    const float4* b4 = (const float4*)bias;
    const float4 bb0 = b4[lane];
    const float4 bb1 = b4[lane + 32];
    float4 o0, o1;
    o0.x = fmaxf(acc0.x + bb0.x, 0.f); o0.y = fmaxf(acc0.y + bb0.y, 0.f);
    o0.z = fmaxf(acc0.z + bb0.z, 0.f); o0.w = fmaxf(acc0.w + bb0.w, 0.f);
    o1.x = fmaxf(acc1.x + bb1.x, 0.f); o1.y = fmaxf(acc1.y + bb1.y, 0.f);
    o1.z = fmaxf(acc1.z + bb1.z, 0.f); o1.w = fmaxf(acc1.w + bb1.w, 0.f);

    float4* op = (float4*)(out + (size_t)node * UNITS);
    op[lane]      = o0;
    op[lane + 32] = o1;
}

// ===========================================================================
// Fallback path (used only if ws_size can't hold the CSR scratch)
// ===========================================================================
__global__ void gcn_init_bias(float* __restrict__ out,
                              const float* __restrict__ bias, int total4) {
    int i = blockIdx.x * blockDim.x + threadIdx.x;
    if (i >= total4) return;
    ((float4*)out)[i] = ((const float4*)bias)[i & 63];
}

__global__ void gcn_spmm_scatter(const int* __restrict__ eRow,
                                 const int* __restrict__ eCol,
                                 const float* __restrict__ eVal,
                                 const float* __restrict__ H,
                                 float* __restrict__ out, int nEdges) {
    const int edge = (blockIdx.x * blockDim.x + threadIdx.x) >> 5;
    const int lane = threadIdx.x & 31;
    if (edge >= nEdges) return;
    const int   r = eRow[edge];
    const int   c = eCol[edge];
    const float v = eVal[edge];
    const float4* hp = (const float4*)(H + (size_t)c * UNITS);
    float*        op = out + (size_t)r * UNITS;
    #pragma unroll
    for (int i = 0; i < 2; ++i) {
        const int q = lane + i * 32;
        float4 hv = hp[q];
        const int base = q * 4;
        atomicAdd(op + base + 0, v * hv.x);
        atomicAdd(op + base + 1, v * hv.y);
        atomicAdd(op + base + 2, v * hv.z);
        atomicAdd(op + base + 3, v * hv.w);
    }
}

__global__ void gcn_relu(float* __restrict__ out, int total4) {
    int i = blockIdx.x * blockDim.x + threadIdx.x;
    if (i >= total4) return;
    float4 v = ((float4*)out)[i];
    v.x = fmaxf(v.x, 0.f); v.y = fmaxf(v.y, 0.f);
    v.z = fmaxf(v.z, 0.f); v.w = fmaxf(v.w, 0.f);
    ((float4*)out)[i] = v;
}

// ===========================================================================
// Launcher. Inputs: edge_row, edge_col, edge_val, x, weight, bias
// ===========================================================================
extern "C" void kernel_launch(void* const* d_in, const int* in_sizes, int n_in,
                              void* d_out, int out_size, void* d_ws, size_t ws_size,
                              hipStream_t stream) {
    const int*   eRow = (const int*)  d_in[0];
    const int*   eCol = (const int*)  d_in[1];
    const float* eVal = (const float*)d_in[2];
    const float* X    = (const float*)d_in[3];
    const float* W    = (const float*)d_in[4];
    const float* bias = (const float*)d_in[5];

    const int nEdges = in_sizes[0];
    const int nNodes = in_sizes[3] / D_FEAT;
    float* out = (float*)d_out;

    // ---- workspace layout ----
    char* p = (char*)d_ws;
    float* H = (float*)p;                 p += (size_t)nNodes * UNITS * sizeof(float);
    int*   deg    = (int*)p;              p += (size_t)nNodes * sizeof(int);
    int*   off    = (int*)p;              p += ((size_t)nNodes + 1) * sizeof(int);
    int*   cursor = (int*)p;              p += (size_t)nNodes * sizeof(int);
    int*   binCol = (int*)p;              p += (size_t)nEdges * sizeof(int);
    float* binVal = (float*)p;            p += (size_t)nEdges * sizeof(float);
    const size_t needCSR = (size_t)(p - (char*)d_ws);
    const bool useCSR = (needCSR <= ws_size);

    const int threads = 256;
    const int eBlocks = (nEdges + threads - 1) / threads;

    // Dense transform
    gcn_gemm_xw<<<(nNodes + 15) / 16, threads, 0, stream>>>(X, W, H, nNodes);

    if (useCSR) {
        const int zn = 2 * nNodes + 1;   // zeroes contiguous deg and off; cursor is not cleared
        gcn_zero_ints<<<(zn + threads - 1) / threads, threads, 0, stream>>>(deg, zn);
        gcn_degree<<<eBlocks, threads, 0, stream>>>(eRow, deg, nEdges);
        gcn_scan<<<1, 1024, 0, stream>>>(deg, off, nNodes);
        gcn_bin<<<eBlocks, threads, 0, stream>>>(eRow, eCol, eVal, off, cursor,
                                                 binCol, binVal, nEdges);
        const long long t = (long long)nNodes * 32;
        gcn_spmm_rows<<<(int)((t + threads - 1) / threads), threads, 0, stream>>>(
            off, binCol, binVal, H, bias, out, nNodes);
    } else {
        const int total4 = nNodes * UNITS / 4;
        const int oBlocks = (total4 + threads - 1) / threads;
        gcn_init_bias<<<oBlocks, threads, 0, stream>>>(out, bias, total4);
        const long long t = (long long)nEdges * 32;
        gcn_spmm_scatter<<<(int)((t + threads - 1) / threads), threads, 0, stream>>>(
            eRow, eCol, eVal, H, out, nEdges);
        gcn_relu<<<oBlocks, threads, 0, stream>>>(out, total4);
    }
}